// SPDNetwork_89266600280692
// MI455X (gfx1250) — compile-verified
//
#include <hip/hip_runtime.h>

typedef __attribute__((ext_vector_type(2))) float v2f;
typedef __attribute__((ext_vector_type(8))) float v8f;

#define CC 62
#define DD 8
#define EPSV 1e-4f
#define NSWEEP 6

// ---------------------------------------------------------------------------
// Kernel 1: one wave per batch element.
//   W = w1^T @ X[b]   (16x64 padded; rows>=8 and cols>=62 exactly zero)
//   Y = W @ w1        (16x16 padded; valid 8x8 block)
// V_WMMA_F32_16X16X4_F32; layouts per ISA 7.12.2.
// X is bitwise symmetric (X = A A^T / C), so B-fragment elements
// B[r][col]=X[r][col]=X[col][r] are loaded as an adjacent, 8B-aligned pair
// from row `col` -> one global_load_b64 with immediate offset per fragment.
// ---------------------------------------------------------------------------
__global__ __launch_bounds__(256) void spd_bimap1_kernel(
    const float* __restrict__ X, const float* __restrict__ w1,
    float* __restrict__ M1)
{
  __shared__ alignas(16) float Wlds[8 * 16 * 66];  // 8 waves * 16 rows * (64+2)
  const int lane = threadIdx.x & 31;
  const int wv   = threadIdx.x >> 5;
  const int b    = blockIdx.x * 8 + wv;
  const int nn   = lane & 15;   // M index (as A) / N index (as B)
  const int half = lane >> 4;   // selects K pair {0,1} or {2,3}

  // w1 fragments; register image serves both (w1^T as A) and (w1 as B).
  v2f wf[16];
#pragma unroll
  for (int t = 0; t < 16; ++t) {
    int r0 = 4 * t + 2 * half;
    wf[t].x = (nn < DD && r0     < CC) ? w1[r0 * DD + nn]       : 0.0f;
    wf[t].y = (nn < DD && (r0+1) < CC) ? w1[(r0 + 1) * DD + nn] : 0.0f;
  }

  const float* Xb = X + (size_t)b * (CC * CC);
  float* wl = Wlds + wv * (16 * 66);
  const v2f vzero = {0.0f, 0.0f};

  // ---- stage 1: W = w1^T @ X  (stream X exactly once) ----
#pragma unroll
  for (int nt = 0; nt < 4; ++nt) {
    int  col = nt * 16 + nn;
    bool cok = col < CC;
    const float* Xrow = Xb + (size_t)col * CC;   // row `col` (symmetry)
    v8f acc = {0.f, 0.f, 0.f, 0.f, 0.f, 0.f, 0.f, 0.f};
#pragma unroll
    for (int t = 0; t < 16; ++t) {
      int r0 = 4 * t + 2 * half;                 // even; pair (r0, r0+1)
      v2f bf = (cok && r0 < CC) ? *(const v2f*)(Xrow + r0) : vzero;
      acc = __builtin_amdgcn_wmma_f32_16x16x4_f32(false, wf[t], false, bf,
                                                  (short)0, acc, false, false);
    }
    // D layout: vgpr i -> W[i, col] (lanes 0-15) / W[i+8, col] (lanes 16-31);
    // rows 8..15 are exact zeros, store unmasked.
#pragma unroll
    for (int i = 0; i < 8; ++i) {
      int row = i + 8 * half;
      wl[row * 66 + nt * 16 + nn] = acc[i];
    }
  }
  __syncthreads();

  // ---- stage 2: Y = W @ w1 ----
  v8f accY = {0.f, 0.f, 0.f, 0.f, 0.f, 0.f, 0.f, 0.f};
#pragma unroll
  for (int t = 0; t < 16; ++t) {
    int r0 = 4 * t + 2 * half;
    v2f af = *(const v2f*)(wl + nn * 66 + r0);   // ds_load_b64, zero-padded W
    accY = __builtin_amdgcn_wmma_f32_16x16x4_f32(false, af, false, wf[t],
                                                 (short)0, accY, false, false);
  }
  if (lane < 8) {
    float* o = M1 + (size_t)b * 64;
#pragma unroll
    for (int i = 0; i < 8; ++i) o[i * 8 + lane] = accY[i];  // Y[i, lane]
  }
}

// ---------------------------------------------------------------------------
// Kernel 2: one lane per batch element. Per-lane 8x8 matrices in LDS with
// stride-65 rows -> bank (tid + idx) % 64, conflict-free across a wave32.
// Fused symmetric Jacobi rotation; segmented k-loops avoid per-k compares;
// relative threshold lets fully-converged waves skip rotations via execz.
// ---------------------------------------------------------------------------
#define JROT_SEG(A, V, k)                                   \
  {                                                         \
    float akp = A[(k) * 8 + p], akq = A[(k) * 8 + q];       \
    float np = c * akp - s * akq;                           \
    float nq = s * akp + c * akq;                           \
    A[(k) * 8 + p] = np;  A[p * 8 + (k)] = np;              \
    A[(k) * 8 + q] = nq;  A[q * 8 + (k)] = nq;              \
  }

static __device__ __forceinline__ void jacobi8(float* A, float* V, float* w)
{
  for (int i = 0; i < 8; ++i)
    for (int j = 0; j < 8; ++j) V[i * 8 + j] = (i == j) ? 1.0f : 0.0f;

  for (int sw = 0; sw < NSWEEP; ++sw) {
    for (int p = 0; p < 7; ++p) {
      for (int q = p + 1; q < 8; ++q) {
        float apq = A[p * 8 + q];
        float app = A[p * 8 + p], aqq = A[q * 8 + q];
        if (fabsf(apq) > 1e-10f * (fabsf(app) + fabsf(aqq))) {
          float tau = (aqq - app) / (2.0f * apq);
          float t = copysignf(1.0f, tau) / (fabsf(tau) + sqrtf(1.0f + tau * tau));
          float c = 1.0f / sqrtf(1.0f + t * t);
          float s = t * c;
          // off-(p,q) elements: symmetric fused update of J^T A J
          for (int k = 0; k < p; ++k)     JROT_SEG(A, V, k)
          for (int k = p + 1; k < q; ++k) JROT_SEG(A, V, k)
          for (int k = q + 1; k < 8; ++k) JROT_SEG(A, V, k)
          A[p * 8 + p] = app - t * apq;          // exact diagonal identities
          A[q * 8 + q] = aqq + t * apq;
          A[p * 8 + q] = 0.0f;  A[q * 8 + p] = 0.0f;
          for (int k = 0; k < 8; ++k) {          // V <- V*J
            float vkp = V[k * 8 + p], vkq = V[k * 8 + q];
            V[k * 8 + p] = c * vkp - s * vkq;
            V[k * 8 + q] = s * vkp + c * vkq;
          }
        }
      }
    }
  }
  for (int k = 0; k < 8; ++k) w[k] = A[k * 8 + k];
}

__global__ __launch_bounds__(32) void spd_eig_kernel(
    const float* __restrict__ M1, const float* __restrict__ w2,
    const float* __restrict__ fc, float* __restrict__ out, int B)
{
  __shared__ float Asm[32 * 65];
  __shared__ float Vsm[32 * 65];
  const int tid = threadIdx.x;
  const int b   = blockIdx.x * 32 + tid;
  float* A = Asm + tid * 65;
  float* V = Vsm + tid * 65;

  // load + symmetrize (WMMA rounding can break exact symmetry)
  const float* M = M1 + (size_t)b * 64;
  for (int i = 0; i < 8; ++i)
    for (int j = 0; j < 8; ++j)
      A[i * 8 + j] = 0.5f * (M[i * 8 + j] + M[j * 8 + i]);

  float w[8];
  jacobi8(A, V, w);                     // eig of M1
  float fw[8];
  for (int k = 0; k < 8; ++k) fw[k] = fmaxf(w[k], EPSV);

  // G = V^T @ w2  -> store into A (A's old contents no longer needed)
  for (int k = 0; k < 8; ++k) {
    float vk[8];
#pragma unroll
    for (int c = 0; c < 8; ++c) vk[c] = V[c * 8 + k];
    float g[8];
#pragma unroll
    for (int a = 0; a < 8; ++a) {
      float sum = 0.0f;
#pragma unroll
      for (int c = 0; c < 8; ++c) sum += vk[c] * w2[c * 8 + a];
      g[a] = sum;
    }
#pragma unroll
    for (int a = 0; a < 8; ++a) A[k * 8 + a] = g[a];
  }
  // M2[a][e] = sum_k fw[k] * G[k][a] * G[k][e]  (= w2^T rec(M1) w2) -> V
  for (int a = 0; a < 8; ++a) {
    float ga[8];
#pragma unroll
    for (int k = 0; k < 8; ++k) ga[k] = fw[k] * A[k * 8 + a];
    for (int e = 0; e < 8; ++e) {
      float sum = 0.0f;
#pragma unroll
      for (int k = 0; k < 8; ++k) sum += ga[k] * A[k * 8 + e];
      V[a * 8 + e] = sum;
    }
  }
  for (int i = 0; i < 64; ++i) A[i] = V[i];   // M2 -> A

  jacobi8(A, V, w);                     // eig of M2; logm(rec(M2)) reuses it
  float lw[8];
  for (int k = 0; k < 8; ++k) lw[k] = logf(fmaxf(w[k], EPSV));

  // L = V diag(lw) V^T ; feat + logits
  float z0 = 0.0f, z1 = 0.0f;
  float* feat = out + (size_t)2 * B + (size_t)b * 64;
  for (int i = 0; i < 8; ++i) {
    float vi[8];
#pragma unroll
    for (int k = 0; k < 8; ++k) vi[k] = V[i * 8 + k] * lw[k];
    for (int j = 0; j < 8; ++j) {
      float sum = 0.0f;
#pragma unroll
      for (int k = 0; k < 8; ++k) sum += vi[k] * V[j * 8 + k];
      int f = i * 8 + j;
      feat[f] = sum;
      z0 += sum * fc[f * 2 + 0];
      z1 += sum * fc[f * 2 + 1];
    }
  }
  float m = fmaxf(z0, z1);
  float l = logf(expf(z0 - m) + expf(z1 - m));
  out[(size_t)b * 2 + 0] = z0 - m - l;
  out[(size_t)b * 2 + 1] = z1 - m - l;
}

extern "C" void kernel_launch(void* const* d_in, const int* in_sizes, int n_in,
                              void* d_out, int out_size, void* d_ws, size_t ws_size,
                              hipStream_t stream) {
  const float* X  = (const float*)d_in[0];   // [B,62,62]
  const float* w1 = (const float*)d_in[1];   // [62,8]
  const float* w2 = (const float*)d_in[2];   // [8,8]
  const float* fc = (const float*)d_in[3];   // [64,2]
  float* out = (float*)d_out;                // [B*2] logprobs ++ [B*64] feat
  float* M1  = (float*)d_ws;                 // [B,8,8] intermediate (2 MB)
  const int B = in_sizes[0] / (CC * CC);     // 8192

  spd_bimap1_kernel<<<B / 8, 256, 0, stream>>>(X, w1, M1);
  spd_eig_kernel<<<B / 32, 32, 0, stream>>>(M1, w2, fc, out, B);
}